// Pack_41927470743575
// MI455X (gfx1250) — compile-verified
//
#include <hip/hip_runtime.h>

typedef float v2f  __attribute__((ext_vector_type(2)));
typedef float v8f  __attribute__((ext_vector_type(8)));
typedef unsigned int u32x4 __attribute__((ext_vector_type(4)));
typedef int   i32x4 __attribute__((ext_vector_type(4)));
typedef int   i32x8 __attribute__((ext_vector_type(8)));

#define B_ROWS   4096
#define N_KEYS   8192
#define DDIM     512
#define VDIM_    512
#define BM       32     // rows of b per workgroup
#define BN       128    // N-chunk
#define DK       64     // D sub-chunk staged in LDS for GEMM1
#define NTHREADS 256    // 8 wave32

// D = A(16x4) * B(4x16) + C, fp32 WMMA
#define WMMA_F32(a, b, c) \
  __builtin_amdgcn_wmma_f32_16x16x4_f32(false, (a), false, (b), (short)0, (c), false, false)

// TDM arity differs across toolchains (ROCm7.2 clang-22: 5 args; clang-23: 6 args)
#if __clang_major__ >= 23
#define TDM_LOAD(g0, g1, z4, z8) \
  __builtin_amdgcn_tensor_load_to_lds((g0), (g1), (z4), (z4), (z8), 0)
#else
#define TDM_LOAD(g0, g1, z4, z8) \
  __builtin_amdgcn_tensor_load_to_lds((g0), (g1), (z4), (z4), 0)
#endif

__global__ __launch_bounds__(NTHREADS, 1)
void idw_wmma_kernel(const float* __restrict__ bq,
                     const float* __restrict__ Kk,
                     const float* __restrict__ Vv,
                     float* __restrict__ out)
{
  __shared__ float sB[BM][DDIM + 4];   // 66,048 B  (pad: lane stride%64banks==4 -> conflict-free)
  __shared__ float sK[BN][DK + 4];     // 34,816 B  (TDM writes this, pad via D# pad fields)
  __shared__ float sW[BM][BN + 4];     // 16,896 B
  __shared__ float sBsq[BM];
  __shared__ float sKsq[BN];
  __shared__ float sSW[BM];

  const int  t       = threadIdx.x;
  const int  wave    = t >> 5;          // 0..7 (wave32)
  const int  lane    = t & 31;
  const int  lh      = lane & 15;
  const bool hi      = lane >= 16;      // lanes 16..31 hold K=2,3 / M=+8 per ISA f32 layout
  const int  mbase   = blockIdx.x * BM;
  const int  colbase = wave * 64;       // 64 output cols per wave (4 WMMA tiles)

  // ---- stage b tile [BM x D] into LDS (coalesced, one-time) ----
  for (int i = t; i < BM * DDIM; i += NTHREADS) {
    int r = i >> 9, c = i & (DDIM - 1);
    sB[r][c] = bq[(size_t)(mbase + r) * DDIM + c];
  }
  if (t < BM) sSW[t] = 0.0f;
  __syncthreads();
  if (t < BM) {                         // per-row |b|^2
    float s = 0.0f;
    for (int c = 0; c < DDIM; ++c) { float x = sB[t][c]; s = fmaf(x, x, s); }
    sBsq[t] = s;
  }

  // ---- TDM descriptor (constant part): 2D tile DK x BN, padded rows ----
  // group1: data_size=2 (4B) | pad_enable | pad_interval=5 (64 DW) | pad_amount=3 (4 DW)
  i32x8 g1;
  g1[0] = (2 << 16) | (1 << 20) | (5 << 22) | (3 << 25);
  g1[1] = (int)((DDIM & 0xFFFF) << 16);                                  // tensor_dim0[15:0] @63:48
  g1[2] = (int)(((DDIM >> 16) & 0xFFFF) | ((N_KEYS & 0xFFFF) << 16));    // dim0 hi | dim1 lo
  g1[3] = (int)(((N_KEYS >> 16) & 0xFFFF) | (DK << 16));                 // dim1 hi | tile_dim0=64
  g1[4] = BN;                                                            // tile_dim1=128, tile_dim2=0
  g1[5] = DDIM;                                                          // tensor_dim0_stride=512
  g1[6] = 0;
  g1[7] = 0;
  const i32x4 z4 = {0, 0, 0, 0};
  const i32x8 z8 = {0, 0, 0, 0, 0, 0, 0, 0};
  const unsigned sk_lds = (unsigned)(size_t)&sK[0][0];                   // LDS byte address
  const unsigned long long kbase = (unsigned long long)(size_t)Kk;

  const v8f vzero = {0.f,0.f,0.f,0.f,0.f,0.f,0.f,0.f};
  v8f acc[2][4];
  #pragma unroll
  for (int tm = 0; tm < 2; ++tm)
    #pragma unroll
    for (int vt = 0; vt < 4; ++vt) acc[tm][vt] = vzero;

  const int ncol = wave * 16 + lh;      // this wave's S-column group

  for (int n0 = 0; n0 < N_KEYS; n0 += BN) {
    // prefetch next K chunk into L2 (global_prefetch_b8)
    {
      int nn = (n0 + BN) & (N_KEYS - 1);
      __builtin_prefetch(Kk + (size_t)(nn + (t >> 1)) * DDIM + (size_t)(t & 1) * 256, 0, 1);
    }

    // ---- GEMM1: S = b . K^T over D, K staged by the Tensor Data Mover ----
    v8f S0 = vzero, S1 = vzero;
    if (t < BN) sKsq[t] = 0.0f;
    for (int d0 = 0; d0 < DDIM; d0 += DK) {
      __syncthreads();                  // all prior readers of sK done
      if (wave == 0) {                  // single DMA per stage (TDM is per-wave, EXEC-ignored)
        const unsigned long long gaddr =
            kbase + (((unsigned long long)n0 * DDIM + (unsigned)d0) << 2);
        u32x4 g0;
        g0.x = 1u;                                    // count=1 (valid), user mode
        g0.y = sk_lds;                                // lds_addr
        g0.z = (unsigned)gaddr;                       // global_addr[31:0]
        g0.w = (unsigned)(gaddr >> 32) | (2u << 30);  // global_addr[56:32] | type=2
        TDM_LOAD(g0, g1, z4, z8);
        __builtin_amdgcn_s_wait_tensorcnt(0);
      }
      __syncthreads();                  // tile visible to all waves
      if (t < BN) {                     // accumulate |k|^2 from the staged slice
        float s = sKsq[t];
        #pragma unroll 8
        for (int c = 0; c < DK; ++c) { float x = sK[t][c]; s = fmaf(x, x, s); }
        sKsq[t] = s;
      }
      #pragma unroll 4
      for (int kk = 0; kk < DK; kk += 4) {
        const int kb = kk + (hi ? 2 : 0);     // K offset per lane half (ISA A/B f32 layout)
        v2f bf; bf.x = sK[ncol][kb];      bf.y = sK[ncol][kb + 1];       // B = K^T frag
        const int ka = d0 + kb;
        v2f a0; a0.x = sB[lh][ka];        a0.y = sB[lh][ka + 1];         // A rows 0..15
        v2f a1; a1.x = sB[16 + lh][ka];   a1.y = sB[16 + lh][ka + 1];    // A rows 16..31
        S0 = WMMA_F32(a0, bf, S0);
        S1 = WMMA_F32(a1, bf, S1);
      }
    }

    // ---- elementwise: w = 1 / max(max(d2,0), 1e-6)  (P=2 => dist^P == clamped d2) ----
    const float ksqn = sKsq[ncol];
    #pragma unroll
    for (int r = 0; r < 8; ++r) {
      {
        int m = r + (hi ? 8 : 0);
        float d2 = fmaxf(sBsq[m] + ksqn - 2.0f * S0[r], 0.0f);
        sW[m][ncol] = 1.0f / fmaxf(d2, 1e-6f);
      }
      {
        int m = 16 + r + (hi ? 8 : 0);
        float d2 = fmaxf(sBsq[m] + ksqn - 2.0f * S1[r], 0.0f);
        sW[m][ncol] = 1.0f / fmaxf(d2, 1e-6f);
      }
    }
    __syncthreads();                    // W complete before sum_w + GEMM2 A-reads

    // ---- sum_w accumulation (ds_add_f32) ----
    {
      int row = t >> 3, seg = t & 7;
      float p = 0.0f;
      #pragma unroll
      for (int j = 0; j < 16; ++j) p += sW[row][seg * 16 + j];
      atomicAdd(&sSW[row], p);
    }

    // ---- GEMM2: acc += W (32xBN) . V (BNx64-per-wave) ----
    #pragma unroll 2
    for (int kk = 0; kk < BN; kk += 4) {
      const int kb = kk + (hi ? 2 : 0);
      v2f a0; a0.x = sW[lh][kb];      a0.y = sW[lh][kb + 1];
      v2f a1; a1.x = sW[16 + lh][kb]; a1.y = sW[16 + lh][kb + 1];
      const float* vp = Vv + (size_t)(n0 + kb) * VDIM_ + colbase + lh;   // coalesced 16-lane rows
      #pragma unroll
      for (int vt = 0; vt < 4; ++vt) {
        v2f bv; bv.x = vp[vt * 16]; bv.y = vp[VDIM_ + vt * 16];
        acc[0][vt] = WMMA_F32(a0, bv, acc[0][vt]);
        acc[1][vt] = WMMA_F32(a1, bv, acc[1][vt]);
      }
    }
    // next-chunk barriers (top of d0 loop) fence sW/sKsq reuse
  }

  // ---- epilogue: divide by sum_w, store ----
  __syncthreads();
  #pragma unroll
  for (int tm = 0; tm < 2; ++tm)
    #pragma unroll
    for (int r = 0; r < 8; ++r) {
      const int m = tm * 16 + r + (hi ? 8 : 0);
      const float inv = 1.0f / sSW[m];
      const size_t base = (size_t)(mbase + m) * VDIM_ + colbase + lh;
      #pragma unroll
      for (int vt = 0; vt < 4; ++vt)
        out[base + vt * 16] = acc[tm][vt][r] * inv;
    }
}

extern "C" void kernel_launch(void* const* d_in, const int* in_sizes, int n_in,
                              void* d_out, int out_size, void* d_ws, size_t ws_size,
                              hipStream_t stream) {
  (void)in_sizes; (void)n_in; (void)d_ws; (void)ws_size; (void)out_size;
  const float* b = (const float*)d_in[0];
  const float* K = (const float*)d_in[1];
  const float* V = (const float*)d_in[2];
  float* out = (float*)d_out;
  idw_wmma_kernel<<<dim3(B_ROWS / BM), dim3(NTHREADS), 0, stream>>>(b, K, V, out);
}